// TransportMapKernel_71494025609608
// MI455X (gfx1250) — compile-verified
//
#include <hip/hip_runtime.h>
#include <math.h>

// Problem constants (from the reference)
#define N_LOCS   30000
#define FIXM     30
#define MAX_M1   31     // augmented_response last-dim
#define XP       34     // xtile LDS pitch (floats): even (8B-aligned v2f) + conflict-free
#define CP       33     // gram/chol LDS pitch: odd -> conflict-free row access
#define WPB      4      // waves (=locations) per block

typedef __attribute__((ext_vector_type(2))) float v2f;
typedef __attribute__((ext_vector_type(4))) float v4f;
typedef __attribute__((ext_vector_type(8))) float v8f;

__device__ __forceinline__ void wave_lds_sync() {
  // intra-wave LDS RAW across lanes: DS ops are in-order per wave; just make
  // sure the compiler doesn't reorder and all DS ops have completed.
  __builtin_amdgcn_wave_barrier();
  asm volatile("s_wait_dscnt 0" ::: "memory");
  __builtin_amdgcn_wave_barrier();
}

__global__ __launch_bounds__(WPB * 32) void tmk_kernel(
    const float* __restrict__ aug,        // (32, N_LOCS, 31)
    const float* __restrict__ scales,     // (N_LOCS)
    const float* __restrict__ nug_mean,   // (N_LOCS)
    const float* __restrict__ frac,       // (1)
    const int*   __restrict__ batch_idx,  // (N_LOCS) int32
    const float* __restrict__ theta_q,    // (1)
    const float* __restrict__ sigma_p,    // (2)
    const float* __restrict__ lengthscale,// (1)
    float* __restrict__ out)              // [g | g_chol | nug_mean]
{
  __shared__ float xtile_s[WPB][32 * XP];
  __shared__ float chol_s [WPB][32 * CP];
  __shared__ float sq_s   [WPB][32];

  const int wid  = threadIdx.x >> 5;
  const int lane = threadIdx.x & 31;
  const int l    = blockIdx.x * WPB + wid;     // wave-uniform location id
  if (l >= N_LOCS) return;                     // uniform per wave; EXEC stays all-1s

  float* xtile = xtile_s[wid];
  float* cholb = chol_s[wid];
  float* sqb   = sq_s[wid];

  // ---- per-location scalars -------------------------------------------------
  const float et     = __expf(theta_q[0]);                            // exp(theta_q)
  const float inv_ls = 1.0f / (__expf(lengthscale[0]) * sqrtf(3.0f)); // ls = e^L*sqrt(2*1.5)
  // sigmas^2 = exp(2*(s0 + s1*log(scale)))
  const float coef   = frac[0] *
      __expf(2.0f * (sigma_p[0] + sigma_p[1] * __logf(scales[l]))) / nug_mean[l];
  const bool  bz     = (batch_idx[l] == 0);

  // ---- Phase 1: load sample row (lane = sample), scale, square-sum ----------
  // scaling_k = exp(-0.5*k*et)/ls is a geometric sequence: one exp, 30 muls.
  {
    const size_t base = ((size_t)lane * (size_t)N_LOCS + (size_t)l) * MAX_M1 + 1;
    const float rfac = __expf(-0.5f * et);
    float fac = inv_ls;
    float ss = 0.0f;
#pragma unroll
    for (int k = 0; k < FIXM; ++k) {
      fac *= rfac;                                          // = exp(-0.5*(k+1)*et)/ls
      float v = aug[base + k];
      v = (v != v) ? 0.0f : v;                              // nan -> 0
      v *= fac;
      xtile[lane * XP + k] = v;
      ss = fmaf(v, v, ss);
    }
    xtile[lane * XP + 30] = 0.0f;                           // pad K 30 -> 32
    xtile[lane * XP + 31] = 0.0f;
    sqb[lane] = ss;
  }
  wave_lds_sync();

  // ---- Phase 2: Gram = xl * xl^T via v_wmma_f32_16x16x4_f32 -----------------
  // For Gram, the A-frag of row-tile t and the B-frag of col-tile t are the
  // same register data: value(lane,v) = xl[16t + (lane&15)][4kc + 2*(lane>>4) + v]
  v8f a00 = {}; v8f a01 = {}; v8f a10 = {}; v8f a11 = {};
  const int m = lane & 15;
  const int h = lane >> 4;
#pragma unroll
  for (int kc = 0; kc < 8; ++kc) {
    const int k0 = 4 * kc + 2 * h;
    v2f f0 = *(const v2f*)&xtile[(     m) * XP + k0];
    v2f f1 = *(const v2f*)&xtile[(16 + m) * XP + k0];
    a00 = __builtin_amdgcn_wmma_f32_16x16x4_f32(false, f0, false, f0, (short)0, a00, false, false);
    a01 = __builtin_amdgcn_wmma_f32_16x16x4_f32(false, f0, false, f1, (short)0, a01, false, false);
    a10 = __builtin_amdgcn_wmma_f32_16x16x4_f32(false, f1, false, f0, (short)0, a10, false, false);
    a11 = __builtin_amdgcn_wmma_f32_16x16x4_f32(false, f1, false, f1, (short)0, a11, false, false);
  }

  // ---- Phase 3: Matern-1.5 elementwise on C/D layout, stash g in LDS --------
  const float sqj0 = sqb[m];
  const float sqj1 = sqb[16 + m];
  auto emit = [&](v8f acc, int ti, int tj, float sqj) {
#pragma unroll
    for (int r = 0; r < 8; ++r) {
      const int i = 16 * ti + 8 * h + r;
      const int j = 16 * tj + m;
      const float sqi = sqb[i];
      const float d2  = fmaxf(fmaf(-2.0f, acc[r], sqi + sqj), 0.0f);
      const float s3d = sqrtf(3.0f * d2);
      float g = coef * (1.0f + s3d) * __expf(-s3d);
      const float diag = (i == j) ? 1.0f : 0.0f;
      g += diag;
      g = bz ? diag : g;
      cholb[i * CP + j] = g;
    }
  };
  emit(a00, 0, 0, sqj0);
  emit(a01, 0, 1, sqj1);
  emit(a10, 1, 0, sqj0);
  emit(a11, 1, 1, sqj1);
  wave_lds_sync();

  // ---- Phase 4: coalesced b128 nontemporal write of g -----------------------
  {
    v4f* __restrict__ og = (v4f*)(out + (size_t)l * 1024);
#pragma unroll
    for (int it = 0; it < 8; ++it) {
      const int i = it * 4 + (lane >> 3);       // row
      const int j = (lane & 7) * 4;             // col
      v4f t;
      t[0] = cholb[i * CP + j + 0];
      t[1] = cholb[i * CP + j + 1];
      t[2] = cholb[i * CP + j + 2];
      t[3] = cholb[i * CP + j + 3];
      __builtin_nontemporal_store(t, og + it * 32 + lane);  // 512B/wave/instr
    }
  }

  // ---- Phase 5: register Cholesky, lane = row, full unroll ------------------
  float r_[32];
#pragma unroll
  for (int j = 0; j < 32; ++j) r_[j] = cholb[lane * CP + j];

#pragma unroll
  for (int k = 0; k < 32; ++k) {
    const float dk  = __shfl(r_[k], k, 32);        // diag from lane k
    const float inv = __builtin_amdgcn_rsqf(dk);   // v_rsq_f32: no divide chain
    const float lkk = dk * inv;                    // sqrt(dk)
    // finalized column-k value for this row; 0 above the diagonal
    float v = (lane == k) ? lkk : ((lane > k) ? r_[k] * inv : 0.0f);
    r_[k] = v;
    // trailing update: v==0 for finalized rows makes this a no-op there;
    // row-k upper junk is overwritten by the v=0 case when its column finalizes.
#pragma unroll
    for (int j = k + 1; j < 32; ++j) {
      const float ljk = __shfl(v, j, 32);          // L[j][k] broadcast
      r_[j] = fmaf(-v, ljk, r_[j]);
    }
  }
  wave_lds_sync();
#pragma unroll
  for (int j = 0; j < 32; ++j) cholb[lane * CP + j] = r_[j];
  wave_lds_sync();

  // ---- Phase 6: coalesced b128 nontemporal write of g_chol + nug_mean -------
  {
    v4f* __restrict__ oc = (v4f*)(out + (size_t)N_LOCS * 1024 + (size_t)l * 1024);
#pragma unroll
    for (int it = 0; it < 8; ++it) {
      const int i = it * 4 + (lane >> 3);
      const int j = (lane & 7) * 4;
      v4f t;
      t[0] = cholb[i * CP + j + 0];
      t[1] = cholb[i * CP + j + 1];
      t[2] = cholb[i * CP + j + 2];
      t[3] = cholb[i * CP + j + 3];
      __builtin_nontemporal_store(t, oc + it * 32 + lane);
    }
  }

  if (lane == 0)
    out[(size_t)2 * N_LOCS * 1024 + (size_t)l] = nug_mean[l];
}

extern "C" void kernel_launch(void* const* d_in, const int* in_sizes, int n_in,
                              void* d_out, int out_size, void* d_ws, size_t ws_size,
                              hipStream_t stream) {
  (void)in_sizes; (void)n_in; (void)out_size; (void)d_ws; (void)ws_size;
  const float* aug       = (const float*)d_in[0];
  const float* scales    = (const float*)d_in[1];
  const float* nug_mean  = (const float*)d_in[2];
  const float* frac      = (const float*)d_in[3];
  const int*   batch_idx = (const int*)  d_in[4];
  const float* theta_q   = (const float*)d_in[5];
  const float* sigma_p   = (const float*)d_in[6];
  const float* lengthsc  = (const float*)d_in[7];
  float* out = (float*)d_out;

  const int grid = (N_LOCS + WPB - 1) / WPB;   // 7500 blocks, 4 waves each
  tmk_kernel<<<grid, WPB * 32, 0, stream>>>(aug, scales, nug_mean, frac, batch_idx,
                                            theta_q, sigma_p, lengthsc, out);
}